// slot_dec_35424890258115
// MI455X (gfx1250) — compile-verified
//
#include <hip/hip_runtime.h>

typedef float v2f __attribute__((ext_vector_type(2)));
typedef float v8f __attribute__((ext_vector_type(8)));

#define HD   256      // H
#define BB   64       // batch
#define LL   128      // seq len
#define LB   128      // label size
#define HID4 1024     // 4H
#define IN5  1280     // 5H
#define ROWS (LL*BB)  // 8192 cells in the serial chain
#define NBLK 16       // workgroups cooperating on the chain
#define UPB  16       // hidden units per block (256/16)

// ---------------------------------------------------------------------------
// Kernel 1: Gx[(t*B+b), j] = sum_{k<1024} feats(t,b,k) * W_ih[j,k] + b_ih[j]+b_hh[j]
// feats(t,b,k) = x[b,t,k] (k<512) else hi[b,t,k-512]
// One wave per 16x16 tile, K-loop of V_WMMA_F32_16X16X4_F32 (full fp32).
// Also resets the chain-barrier counter (stream-ordered before lstm_scan).
// ---------------------------------------------------------------------------
__global__ __launch_bounds__(256) void gx_gemm(
    const float* __restrict__ x, const float* __restrict__ hi,
    const float* __restrict__ W_ih,
    const float* __restrict__ b_ih, const float* __restrict__ b_hh,
    float* __restrict__ Gx, unsigned* __restrict__ cnt)
{
    if (blockIdx.x == 0 && threadIdx.x == 0) *cnt = 0u;

    const int wave = (blockIdx.x * blockDim.x + threadIdx.x) >> 5; // 0..32767
    const int lane = threadIdx.x & 31;
    const int mt = wave >> 6;           // 0..511  M-tile (rows t*64+b)
    const int nt = wave & 63;           // 0..63   N-tile (gate cols)
    const int t  = mt >> 2;             // 4 M-tiles per timestep
    const int b0 = (mt & 3) << 4;
    const int j0 = nt << 4;
    const int half = lane >> 4;         // 0: K0/K1, 1: K2/K3
    const int ln   = lane & 15;         // M (for A) / N (for B)

    const int    frow = ((b0 + ln) * LL + t) * (2 * HD); // base into x / hi
    const float* wr   = W_ih + (size_t)(j0 + ln) * IN5;

    v8f acc = {};
    for (int k0 = 0; k0 < HID4; k0 += 4) {
        const int kk = k0 + 2 * half;   // never straddles the 512 boundary
        v2f a, w;
        if (kk < 2 * HD) { a.x = x[frow + kk];            a.y = x[frow + kk + 1]; }
        else             { a.x = hi[frow + kk - 2 * HD];  a.y = hi[frow + kk - 2 * HD + 1]; }
        w.x = wr[kk]; w.y = wr[kk + 1];
        acc = __builtin_amdgcn_wmma_f32_16x16x4_f32(false, a, false, w,
                                                    (short)0, acc, false, false);
    }

    const float bias = b_ih[j0 + ln] + b_hh[j0 + ln];
    float* out = Gx + ((size_t)(t * BB + b0 + 8 * half)) * HID4 + j0 + ln;
#pragma unroll
    for (int r = 0; r < 8; ++r)
        out[(size_t)r * HID4] = acc[r] + bias;
}

// ---------------------------------------------------------------------------
// Kernel 2: 8192-cell serial LSTM chain across NBLK cooperating workgroups.
// Block blk owns hidden units [blk*16, blk*16+16) -> 64 gate rows, K=512,
// weights held entirely in VGPRs (32 x float4 per thread, 4 threads/row).
// h history lives in Hbuf: chain-h row = cell-1, out_prev row = cell-64.
// Cross-WGP sync: fence + monotonic atomic counter + s_sleep spin.
// ---------------------------------------------------------------------------
__device__ __forceinline__ float sigf(float v) {
    return 1.0f / (1.0f + __expf(-v));
}
__device__ __forceinline__ float tanhfast(float v) {
    float e = __expf(2.0f * v);
    return (e - 1.0f) / (e + 1.0f);
}

__global__ __launch_bounds__(256, 1) void lstm_scan(
    const float* __restrict__ Gx,
    const float* __restrict__ W_ih,
    const float* __restrict__ W_hh,
    float* Hbuf,                      // read + written here: no restrict
    unsigned* cnt)
{
    __shared__ float garr[64];

    const int tid  = threadIdx.x;     // 0..255
    const int blk  = blockIdx.x;      // 0..15
    const int r    = tid >> 2;        // 0..63 : row = gate*16 + uu
    const int q    = tid & 3;         // K-quarter (128 elements)
    const int uu   = r & 15;
    const int gate = r >> 4;
    const int j    = gate * HD + blk * UPB + uu;   // global gate row (0..1023)

    // ---- preload this thread's 128 weights into registers ----
    // K layout: k<256 -> W_hh[j][k] ; k>=256 -> W_ih[j][1024 + (k-256)]
    const float* wsrc = (q < 2) ? (W_hh + (size_t)j * HD + q * 128)
                                : (W_ih + (size_t)j * IN5 + HID4 + (q - 2) * 128);
    float4 w[32];
#pragma unroll
    for (int k = 0; k < 32; ++k) w[k] = ((const float4*)wsrc)[k];

    float c = 0.0f;                   // cell state, live in threads tid<16

    for (int cell = 0; cell < ROWS; ++cell) {
        // prefetch next cell's Gx slice into cache while we compute
        if (tid < UPB)
            __builtin_prefetch(Gx + (size_t)(cell + 1) * HID4 + blk * UPB + tid, 0, 1);

        // hv slice: q<2 -> chain h (row cell-1), q>=2 -> out_prev (row cell-64)
        const int hrow = (q < 2) ? (cell - 1) : (cell - BB);
        const float4* hsrc =
            (const float4*)(Hbuf + (size_t)hrow * HD + (q & 1) * 128);
        float acc = 0.0f;
        if (hrow >= 0) {
#pragma unroll
            for (int k = 0; k < 32; ++k) {
                const float4 hv = hsrc[k];
                acc += w[k].x * hv.x + w[k].y * hv.y + w[k].z * hv.z + w[k].w * hv.w;
            }
        }
        // 4-way reduce within the quad (threads 4r..4r+3, same wave32)
        acc += __shfl_xor(acc, 1, 32);
        acc += __shfl_xor(acc, 2, 32);
        if (q == 0) garr[r] = acc;
        __syncthreads();

        if (tid < UPB) {
            const float* gx = Gx + (size_t)cell * HID4 + blk * UPB + tid;
            const float gi = garr[tid]      + gx[0];
            const float gf = garr[16 + tid] + gx[256];
            const float gc = garr[32 + tid] + gx[512];
            const float go = garr[48 + tid] + gx[768];
            const float i = sigf(gi);
            const float f = sigf(gf);
            const float g = tanhfast(gc);
            const float o = sigf(go);
            c = f * c + i * g;
            const float hn = o * tanhfast(c);
            Hbuf[(size_t)cell * HD + blk * UPB + tid] = hn;
        }
        __threadfence();              // release: publish hn device-wide
        __syncthreads();
        if (tid == 0) {
            __hip_atomic_fetch_add(cnt, 1u, __ATOMIC_RELEASE,
                                   __HIP_MEMORY_SCOPE_AGENT);
            const unsigned target = (unsigned)(NBLK * (cell + 1));
            while (__hip_atomic_load(cnt, __ATOMIC_ACQUIRE,
                                     __HIP_MEMORY_SCOPE_AGENT) < target)
                __builtin_amdgcn_s_sleep(1);
        }
        __syncthreads();
        __threadfence();              // acquire: invalidate before reading peers' h
    }
}

// ---------------------------------------------------------------------------
// Kernel 3: out[b,l,m] = sum_k Hbuf[(l*B+b), k] * W_fc[m,k] + b_fc[m]
// (handles the (L,B,H) -> (B,L,H) transpose via row indexing)
// ---------------------------------------------------------------------------
__global__ __launch_bounds__(256) void fc_gemm(
    const float* __restrict__ Hbuf, const float* __restrict__ W_fc,
    const float* __restrict__ b_fc, float* __restrict__ out)
{
    const int wave = (blockIdx.x * blockDim.x + threadIdx.x) >> 5; // 0..4095
    const int lane = threadIdx.x & 31;
    const int mt = wave >> 3;          // 0..511: (b, l-tile)
    const int nt = wave & 7;           // 0..7
    const int b  = mt >> 3;            // 0..63
    const int l0 = (mt & 7) << 4;
    const int n0 = nt << 4;
    const int half = lane >> 4;
    const int ln   = lane & 15;

    const float* ar = Hbuf + ((size_t)(l0 + ln) * BB + b) * HD; // A row i=ln
    const float* wr = W_fc + (size_t)(n0 + ln) * HD;            // B col n=ln

    v8f acc = {};
    for (int k0 = 0; k0 < HD; k0 += 4) {
        const int kk = k0 + 2 * half;
        v2f a, w;
        a.x = ar[kk]; a.y = ar[kk + 1];
        w.x = wr[kk]; w.y = wr[kk + 1];
        acc = __builtin_amdgcn_wmma_f32_16x16x4_f32(false, a, false, w,
                                                    (short)0, acc, false, false);
    }

    const float bias = b_fc[n0 + ln];
    float* orow = out + ((size_t)b * LL + l0 + 8 * half) * LB + n0 + ln;
#pragma unroll
    for (int r = 0; r < 8; ++r)
        orow[(size_t)r * LB] = acc[r] + bias;
}

// ---------------------------------------------------------------------------
extern "C" void kernel_launch(void* const* d_in, const int* in_sizes, int n_in,
                              void* d_out, int out_size, void* d_ws, size_t ws_size,
                              hipStream_t stream)
{
    (void)in_sizes; (void)n_in; (void)out_size; (void)ws_size;
    const float* x    = (const float*)d_in[0];
    const float* hi   = (const float*)d_in[1];
    const float* W_ih = (const float*)d_in[2];
    const float* W_hh = (const float*)d_in[3];
    const float* b_ih = (const float*)d_in[4];
    const float* b_hh = (const float*)d_in[5];
    const float* W_fc = (const float*)d_in[6];
    const float* b_fc = (const float*)d_in[7];
    float* out = (float*)d_out;

    float*    Gx   = (float*)d_ws;                   // 8192 x 1024 f32 (32 MB)
    float*    Hbuf = Gx + (size_t)ROWS * HID4;       // 8192 x 256  f32 (8 MB)
    unsigned* cnt  = (unsigned*)(Hbuf + (size_t)ROWS * HD);

    // 512 M-tiles x 64 N-tiles = 32768 waves, 8 waves/block (also zeroes cnt)
    gx_gemm<<<4096, 256, 0, stream>>>(x, hi, W_ih, b_ih, b_hh, Gx, cnt);
    // the serial 8192-cell chain: 16 cooperating workgroups
    lstm_scan<<<NBLK, 256, 0, stream>>>(Gx, W_ih, W_hh, Hbuf, cnt);
    // 512 x 8 = 4096 waves, 8 waves/block
    fc_gemm<<<512, 256, 0, stream>>>(Hbuf, W_fc, b_fc, out);
}